// GraphConvLayer_70016556859563
// MI455X (gfx1250) — compile-verified
//
#include <hip/hip_runtime.h>

// Problem constants (from the reference).
#define S_IN   8192      // input size (x)
#define OUT_N  8192      // output size
#define NNZ_N  524288    // sparse triples (multiple of 4)
#define WNUM_N 30000     // shared weight pool (multiple of 4)
#define BNUM_N 100       // shared bias pool

// ---------------------------------------------------------------------------
// Kernel 1: fully (re)initialize the output with the gathered bias.
// out[col] = Param_b[bias_pos[col]]  -- overwrites the whole poisoned buffer.
// ---------------------------------------------------------------------------
__global__ __launch_bounds__(256) void gc_bias_init(
    const float* __restrict__ Param_b,
    const int*   __restrict__ bias_pos,
    float*       __restrict__ out) {
  int i = blockIdx.x * blockDim.x + threadIdx.x;
  if (i < OUT_N) out[i] = Param_b[bias_pos[i]];
}

// ---------------------------------------------------------------------------
// Kernel 2: sparse scatter-accumulate.
// Stages x (32 KB) and Param_W (117.2 KB) into LDS with CDNA5 async
// global->LDS copies (ASYNCcnt), then streams the triples with int4
// (b128) coalesced loads, LDS gathers, and one global fp32 atomic add
// per nonzero.
// ---------------------------------------------------------------------------
__global__ __launch_bounds__(256) void gc_scatter(
    const float* __restrict__ x,
    const float* __restrict__ Param_W,
    const int*   __restrict__ rows,
    const int*   __restrict__ cols,
    const int*   __restrict__ wpos,
    float*       __restrict__ out) {
  __shared__ float xs[S_IN];      //  32768 B
  __shared__ float ws[WNUM_N];    // 120000 B  -> 152768 B total (<160 KB, 2 WGs/WGP)

  const int t = threadIdx.x;

  // --- Async stage x: 8192 floats = 2048 B128 chunks -----------------------
  for (int k = t; k < S_IN / 4; k += 256) {
    unsigned lds_off = (unsigned)(size_t)(&xs[4 * k]);   // addr[31:0] == LDS byte offset
    const float* gptr = x + 4 * k;
    asm volatile("global_load_async_to_lds_b128 %0, %1, off"
                 :: "v"(lds_off), "v"(gptr)
                 : "memory");
  }
  // --- Async stage Param_W: 30000 floats = 7500 B128 chunks ----------------
  for (int k = t; k < WNUM_N / 4; k += 256) {
    unsigned lds_off = (unsigned)(size_t)(&ws[4 * k]);
    const float* gptr = Param_W + 4 * k;
    asm volatile("global_load_async_to_lds_b128 %0, %1, off"
                 :: "v"(lds_off), "v"(gptr)
                 : "memory");
  }
  // Wait for all async copies, then make LDS visible to the workgroup.
  asm volatile("s_wait_asynccnt 0" ::: "memory");
  __syncthreads();

  // --- Main loop: grid-stride over the triples, 4 items per step -----------
  // Per 4 nonzeros: 3 coalesced b128 index loads + 8 LDS gathers +
  // 4 global_atomic_add_f32 (no return -> STOREcnt).
  const int tid    = blockIdx.x * blockDim.x + t;
  const int stride = gridDim.x * blockDim.x;
  for (int base = tid * 4; base < NNZ_N; base += stride * 4) {
    int4 r4 = *(const int4*)(rows + base);
    int4 c4 = *(const int4*)(cols + base);
    int4 w4 = *(const int4*)(wpos + base);

    float v0 = xs[r4.x] * ws[w4.x];
    float v1 = xs[r4.y] * ws[w4.y];
    float v2 = xs[r4.z] * ws[w4.z];
    float v3 = xs[r4.w] * ws[w4.w];

    atomicAdd(out + c4.x, v0);
    atomicAdd(out + c4.y, v1);
    atomicAdd(out + c4.z, v2);
    atomicAdd(out + c4.w, v3);
  }
}

// ---------------------------------------------------------------------------
// Launcher. Input order: x, Param_W, Param_b, rows, cols, wpos, bias_pos.
// ---------------------------------------------------------------------------
extern "C" void kernel_launch(void* const* d_in, const int* in_sizes, int n_in,
                              void* d_out, int out_size, void* d_ws, size_t ws_size,
                              hipStream_t stream) {
  (void)in_sizes; (void)n_in; (void)out_size; (void)d_ws; (void)ws_size;

  const float* x        = (const float*)d_in[0];
  const float* Param_W  = (const float*)d_in[1];
  const float* Param_b  = (const float*)d_in[2];
  const int*   rows     = (const int*)  d_in[3];
  const int*   cols     = (const int*)  d_in[4];
  const int*   wpos     = (const int*)  d_in[5];
  const int*   bias_pos = (const int*)  d_in[6];
  float*       out      = (float*)d_out;

  // 1) out = gathered bias (full overwrite of poisoned buffer).
  gc_bias_init<<<OUT_N / 256, 256, 0, stream>>>(Param_b, bias_pos, out);

  // 2) scatter-accumulate: 128 WGs x 256 threads, 16 items (4 int4) / thread.
  gc_scatter<<<128, 256, 0, stream>>>(x, Param_W, rows, cols, wpos, out);
}